// ActivationGatedGCNIsotrophicLayer_83476984365537
// MI455X (gfx1250) — compile-verified
//
#include <hip/hip_runtime.h>

#define DIMS 256
#define BN_EPS 1e-5f

typedef float v2f __attribute__((ext_vector_type(2)));
typedef float v4f __attribute__((ext_vector_type(4)));
typedef float v8f __attribute__((ext_vector_type(8)));

// ---------------------------------------------------------------- kernels ---

__global__ void zero_stats_k(float* stats) {
    int i = blockIdx.x * blockDim.x + threadIdx.x;
    if (i < 2 * DIMS) stats[i] = 0.0f;
}

// e passthrough: dominant HBM traffic (1.64 GB round trip). Use non-temporal
// hints so this streaming copy does NOT evict the L2-resident h/agg working
// set (2 x 51.2 MB inside the 192 MB L2) that the gather/scatter phase needs.
// v4f (clang ext vector) is required: the nontemporal builtins reject the
// HIP_vector_type struct.
__global__ void copy_e_k(const v4f* __restrict__ e, v4f* __restrict__ out, long n4) {
    long i = (long)blockIdx.x * blockDim.x + threadIdx.x;
    long stride = (long)gridDim.x * blockDim.x;
    for (; i < n4; i += stride) {
        v4f v = __builtin_nontemporal_load(&e[i]);
        __builtin_nontemporal_store(v, &out[i]);
    }
}

// agg[i,:] = h[i,:] * norm[i]   (self term of aggregation)
__global__ void init_agg_k(const v4f* __restrict__ h, const float* __restrict__ norm,
                           v4f* __restrict__ agg, int n4total) {
    int i = blockIdx.x * blockDim.x + threadIdx.x;
    if (i >= n4total) return;
    int row = i / (DIMS / 4);
    float s = norm[row];
    v4f v = h[i];
    agg[i] = v * s;
}

// scatter-add: 4 edges per 256-thread block; 64 lanes * float4 = one 1KB row.
// agg (51.2MB) and h (51.2MB) both fit in the 192MB L2, so atomics are L2-local.
__global__ void scatter_k(const v4f* __restrict__ h4, const float* __restrict__ norm,
                          const int* __restrict__ src, const int* __restrict__ dst,
                          float* __restrict__ agg, int E) {
    int t = threadIdx.x;
    int e = blockIdx.x * 4 + (t >> 6);
    if (e >= E) return;
    int s = src[e];
    int d = dst[e];
    float sc = norm[s];
    int q = t & 63;  // float4 index within the row
    v4f v = h4[(long)s * (DIMS / 4) + q];
    float* base = agg + (long)d * DIMS + q * 4;
    atomicAdd(base + 0, v.x * sc);
    atomicAdd(base + 1, v.y * sc);
    atomicAdd(base + 2, v.z * sc);
    atomicAdd(base + 3, v.w * sc);
}

// hh = agg * norm   (post-aggregation scale, in place)
__global__ void scale_k(v4f* __restrict__ agg, const float* __restrict__ norm, int n4total) {
    int i = blockIdx.x * blockDim.x + threadIdx.x;
    if (i >= n4total) return;
    int row = i / (DIMS / 4);
    float s = norm[row];
    v4f v = agg[i];
    agg[i] = v * s;
}

// BN statistics via V_WMMA_F32_16X16X4_F32 with A = ones(16x4):
//   D[m,n] = sum_k B[k,n] + C[m,n]  -> exact fp32 column sums of 4x16 tiles.
// 16 waves per block, wave w owns columns [16w, 16w+16); each wave sweeps a
// row partition 4 rows per WMMA pair (sum + sum-of-squares) accumulated in C.
// Main loop is branchless (EXEC stays all-1s, as WMMA requires); only the
// final partial chunk takes the guarded tail path.
// C/D layout: VGPR0 lanes 0-15 = (M=0, N=lane) -> acc[0] holds the colsums.
__global__ void stats_wmma_k(const float* __restrict__ hh, float* __restrict__ colsum,
                             float* __restrict__ colsumsq, int N, int rows_per_block) {
    int lane = threadIdx.x & 31;
    int wave = threadIdx.x >> 5;     // 0..15
    int col  = wave * 16;
    int n    = lane & 15;
    int kh   = lane >> 4;            // which half-wave -> which k rows we feed
    int r0 = blockIdx.x * rows_per_block;
    int r1 = r0 + rows_per_block;
    if (r1 > N) r1 = N;

    v2f ones; ones[0] = 1.0f; ones[1] = 1.0f;
    v8f acc  = {};
    v8f acc2 = {};

    // lane-private streaming pointers: element (r+kh, col+n) and (r+2+kh, col+n)
    const float* pa = hh + (long)(r0 + kh) * DIMS + col + n;
    const float* pb = pa + 2 * DIMS;

    int span  = r1 - r0;
    int rmain = r0 + (span & ~3);    // fully in-range multiple-of-4 portion

    int r = r0;
#pragma unroll 4
    for (; r < rmain; r += 4) {
        float xa = pa[0];
        float xb = pb[0];
        pa += 4 * DIMS;
        pb += 4 * DIMS;
        v2f b;  b[0]  = xa;      b[1]  = xb;
        v2f b2; b2[0] = xa * xa; b2[1] = xb * xb;
        // 8 args: (neg_a, A, neg_b, B, c_mod, C, reuse_a, reuse_b)
        acc  = __builtin_amdgcn_wmma_f32_16x16x4_f32(false, ones, false, b,
                                                     (short)0, acc,  false, false);
        acc2 = __builtin_amdgcn_wmma_f32_16x16x4_f32(false, ones, false, b2,
                                                     (short)0, acc2, false, false);
    }
    if (r < r1) {                    // guarded tail (at most one chunk)
        int ra = r + kh;
        int rb = r + 2 + kh;
        float xa = (ra < r1) ? pa[0] : 0.0f;
        float xb = (rb < r1) ? pb[0] : 0.0f;
        v2f b;  b[0]  = xa;      b[1]  = xb;
        v2f b2; b2[0] = xa * xa; b2[1] = xb * xb;
        acc  = __builtin_amdgcn_wmma_f32_16x16x4_f32(false, ones, false, b,
                                                     (short)0, acc,  false, false);
        acc2 = __builtin_amdgcn_wmma_f32_16x16x4_f32(false, ones, false, b2,
                                                     (short)0, acc2, false, false);
    }

    if (lane < 16) {  // row M=0 of D: lanes 0-15, element 0 = column sums
        atomicAdd(&colsum[col + n],   acc[0]);
        atomicAdd(&colsumsq[col + n], acc2[0]);
    }
}

// BatchNorm (biased var) + affine + ReLU, in place. hh is never re-read after
// this, so write it back non-temporally.
__global__ void finalize_k(float* __restrict__ hh, const float* __restrict__ colsum,
                           const float* __restrict__ colsumsq,
                           const float* __restrict__ gamma, const float* __restrict__ beta,
                           int N, int total) {
    int i = blockIdx.x * blockDim.x + threadIdx.x;
    if (i >= total) return;
    int d = i & (DIMS - 1);
    float invN = 1.0f / (float)N;
    float mean = colsum[d] * invN;
    float var  = colsumsq[d] * invN - mean * mean;
    float inv  = rsqrtf(var + BN_EPS);
    float y = (hh[i] - mean) * inv * gamma[d] + beta[d];
    __builtin_nontemporal_store(fmaxf(y, 0.0f), &hh[i]);
}

// ----------------------------------------------------------------- launch ---

extern "C" void kernel_launch(void* const* d_in, const int* in_sizes, int n_in,
                              void* d_out, int out_size, void* d_ws, size_t ws_size,
                              hipStream_t stream) {
    const float* h     = (const float*)d_in[0];
    const float* e     = (const float*)d_in[1];
    const float* norm  = (const float*)d_in[2];
    const int*   src   = (const int*)d_in[3];
    const int*   dst   = (const int*)d_in[4];
    const float* gamma = (const float*)d_in[5];
    const float* beta  = (const float*)d_in[6];

    const int N = in_sizes[2];          // norm has N elements
    const int E = in_sizes[3];          // src has E elements

    float* hh    = (float*)d_out;                 // [N, 256]
    float* out_e = hh + (size_t)N * DIMS;         // [E, 256] passthrough
    float* stats = (float*)d_ws;                  // [2*256]: colsum, colsumsq
    float* colsum   = stats;
    float* colsumsq = stats + DIMS;

    // 1) zero the stats accumulators (ws is poisoned, not re-zeroed by harness)
    zero_stats_k<<<2, 256, 0, stream>>>(stats);

    // 2) e passthrough (dominant HBM traffic, non-temporal to protect L2)
    long e4 = (long)E * (DIMS / 4);
    copy_e_k<<<8192, 256, 0, stream>>>((const v4f*)e, (v4f*)out_e, e4);

    // 3) agg = h * norm  (self term)
    int n4total = N * (DIMS / 4);
    int blocks_n4 = (n4total + 255) / 256;
    init_agg_k<<<blocks_n4, 256, 0, stream>>>((const v4f*)h, norm, (v4f*)hh, n4total);

    // 4) scatter-add messages into agg (L2-resident atomics)
    int blocks_sc = (E + 3) / 4;
    scatter_k<<<blocks_sc, 256, 0, stream>>>((const v4f*)h, norm, src, dst, hh, E);

    // 5) hh = agg * norm
    scale_k<<<blocks_n4, 256, 0, stream>>>((v4f*)hh, norm, n4total);

    // 6) column sums / sums-of-squares via fp32 WMMA (A = ones trick)
    const int STAT_BLOCKS = 128;
    int rpb = ((N + STAT_BLOCKS * 4 - 1) / (STAT_BLOCKS * 4)) * 4;  // multiple of 4
    stats_wmma_k<<<STAT_BLOCKS, 512, 0, stream>>>(hh, colsum, colsumsq, N, rpb);

    // 7) BN + affine + ReLU in place
    int total = N * DIMS;
    int blocks_fin = (total + 255) / 256;
    finalize_k<<<blocks_fin, 256, 0, stream>>>(hh, colsum, colsumsq, gamma, beta, N, total);
}